// SSIM_loss_80951543595045
// MI455X (gfx1250) — compile-verified
//
#include <hip/hip_runtime.h>

typedef float v2f __attribute__((ext_vector_type(2)));
typedef float v8f __attribute__((ext_vector_type(8)));

#define IMG_H 2048
#define IMG_W 2048
#define OUT_H (IMG_H - 10)   // 2038
#define OUT_W (IMG_W - 10)   // 2038
#define TILE  64
#define P_ROWS 80            // 64 + 16 (halo 10, padded to WMMA row-tiles)
#define P_COLS 76            // 64 + 12 (halo 10 + K-chunk overread pad)
#define H_ROWS 80
#define H_STR  66            // 64 cols + 2 pad (bank decorrelation)
#define NQ 5                 // x, y, xx, yy, xy

__global__ void ssim_init_kernel(float* out) { out[0] = 0.0f; }

__global__ void ssim_final_kernel(float* out) {
    double n = 3.0 * (double)OUT_H * (double)OUT_W;
    out[0] = 1.0f - (float)((double)out[0] / n);
}

__device__ __forceinline__ v8f wmma_f32(v2f a, v2f b, v8f c) {
    return __builtin_amdgcn_wmma_f32_16x16x4_f32(
        false, a, false, b, (short)0, c, false, false);
}

__global__ __launch_bounds__(256)
void ssim_main_kernel(const float* __restrict__ xin,
                      const float* __restrict__ yin,
                      float* __restrict__ accum)
{
    extern __shared__ float lds[];
    float* sX = lds;                          // P_ROWS * P_COLS
    float* sY = sX + P_ROWS * P_COLS;         // P_ROWS * P_COLS
    float* sH = sY + P_ROWS * P_COLS;         // NQ * H_ROWS * H_STR

    const int tid  = threadIdx.x;
    const int lane = tid & 31;
    const int wave = __builtin_amdgcn_readfirstlane(tid >> 5);
    const int mrow = lane & 15;               // M (or N) index within 16
    const int koff = (lane >> 4) << 1;        // K sub-offset per half-wave

    // --- Gaussian weights (size 11, sigma 1.5), normalized ---
    float w[11];
    float wsum = 0.0f;
    #pragma unroll
    for (int i = 0; i < 11; ++i) {
        float d = (float)(i - 5);
        w[i] = __expf(-(d * d) / 4.5f);
        wsum += w[i];
    }
    #pragma unroll
    for (int i = 0; i < 11; ++i) w[i] /= wsum;

    // Banded filter matrix, 28x16 band w[j-c], split into 7 K=4 chunks.
    // Per-lane image is identical for A(16x4: M=lane%16, K=koff+v) and
    // B(4x16: N=lane%16, K=koff+v) layouts.
    v2f wc[7];
    #pragma unroll
    for (int q = 0; q < 7; ++q) {
        #pragma unroll
        for (int v = 0; v < 2; ++v) {
            int K = 4 * q + koff + v;
            int d = K - mrow;
            wc[q][v] = (d >= 0 && d < 11) ? w[d] : 0.0f;
        }
    }

    const int tr0 = blockIdx.y * TILE;
    const int tc0 = blockIdx.x * TILE;
    const float* xc = xin + (size_t)blockIdx.z * ((size_t)IMG_H * IMG_W);
    const float* yc = yin + (size_t)blockIdx.z * ((size_t)IMG_H * IMG_W);

    // --- Load 80x76 patch of x and y into LDS (zero-padded at image edge) ---
    for (int i = tid; i < P_ROWS * P_COLS; i += 256) {
        int r = i / P_COLS;
        int c = i - r * P_COLS;
        int gr = tr0 + r, gc = tc0 + c;
        float xv = 0.0f, yv = 0.0f;
        if (gr < IMG_H && gc < IMG_W) {
            size_t gi = (size_t)gr * IMG_W + gc;
            xv = xc[gi];
            yv = yc[gi];
        }
        sX[i] = xv;
        sY[i] = yv;
    }
    __syncthreads();

    // --- Horizontal pass: 5 row-tiles x 4 col-tiles = 20 jobs; each job
    // computes ALL 5 fields (one ds_load_b64 pair feeds 5 WMMAs per chunk).
    for (int job0 = wave; job0 < 20; job0 += 8) {
        const int job = __builtin_amdgcn_readfirstlane(job0);
        const int rt  = job >> 2;
        const int ct  = job & 3;
        v8f zero = {0.f, 0.f, 0.f, 0.f, 0.f, 0.f, 0.f, 0.f};
        v8f acc[NQ];
        #pragma unroll
        for (int q = 0; q < NQ; ++q) acc[q] = zero;

        const int arow = rt * 16 + mrow;
        #pragma unroll
        for (int ch = 0; ch < 7; ++ch) {
            int acol = ct * 16 + 4 * ch + koff;
            v2f ax = *(const v2f*)(sX + arow * P_COLS + acol);
            v2f ay = *(const v2f*)(sY + arow * P_COLS + acol);
            acc[0] = wmma_f32(ax,      wc[ch], acc[0]);
            acc[1] = wmma_f32(ay,      wc[ch], acc[1]);
            acc[2] = wmma_f32(ax * ax, wc[ch], acc[2]);
            acc[3] = wmma_f32(ay * ay, wc[ch], acc[3]);
            acc[4] = wmma_f32(ax * ay, wc[ch], acc[4]);
        }

        const int hcol  = ct * 16 + mrow;
        const int hrow0 = rt * 16 + ((lane >> 4) << 3);
        #pragma unroll
        for (int q = 0; q < NQ; ++q) {
            float* Hq = sH + q * (H_ROWS * H_STR);
            #pragma unroll
            for (int v = 0; v < 8; ++v)
                Hq[(hrow0 + v) * H_STR + hcol] = acc[q][v];
        }
    }
    __syncthreads();

    // --- Vertical pass + SSIM: 16 output tiles, one wave owns all 5 fields ---
    const float C1 = 1.0e-4f;   // (0.01*1)^2
    const float C2 = 9.0e-4f;   // (0.03*1)^2
    float tsum = 0.0f;
    for (int t0 = wave; t0 < 16; t0 += 8) {
        const int t   = __builtin_amdgcn_readfirstlane(t0);
        const int r0t = (t >> 2) << 4;
        const int c0t = (t & 3) << 4;
        v8f zero = {0.f, 0.f, 0.f, 0.f, 0.f, 0.f, 0.f, 0.f};
        v8f m[NQ];
        #pragma unroll
        for (int q = 0; q < NQ; ++q) m[q] = zero;

        #pragma unroll
        for (int ch = 0; ch < 7; ++ch) {
            int kr = r0t + 4 * ch + koff;
            #pragma unroll
            for (int q = 0; q < NQ; ++q) {
                const float* Hq = sH + q * (H_ROWS * H_STR);
                v2f b;
                b[0] = Hq[(kr + 0) * H_STR + c0t + mrow];
                b[1] = Hq[(kr + 1) * H_STR + c0t + mrow];
                m[q] = wmma_f32(wc[ch], b, m[q]);
            }
        }

        #pragma unroll
        for (int v = 0; v < 8; ++v) {
            int oy = tr0 + r0t + ((lane >> 4) << 3) + v;
            int ox = tc0 + c0t + mrow;
            float mu1 = m[0][v], mu2 = m[1][v];
            float mu1s = mu1 * mu1, mu2s = mu2 * mu2, m12 = mu1 * mu2;
            float s1  = m[2][v] - mu1s;
            float s2  = m[3][v] - mu2s;
            float s12 = m[4][v] - m12;
            float cs  = (2.0f * s12 + C2) / (s1 + s2 + C2);
            float lum = (2.0f * m12 + C1) / (mu1s + mu2s + C1);
            if (oy < OUT_H && ox < OUT_W) tsum += lum * cs;
        }
    }

    // --- Block reduction (reuse sX region) + one atomic per workgroup ---
    float* red = sX;
    red[tid] = tsum;
    __syncthreads();
    for (int off = 128; off > 0; off >>= 1) {
        if (tid < off) red[tid] += red[tid + off];
        __syncthreads();
    }
    if (tid == 0) atomicAdd(accum, red[0]);
}

extern "C" void kernel_launch(void* const* d_in, const int* in_sizes, int n_in,
                              void* d_out, int out_size, void* d_ws, size_t ws_size,
                              hipStream_t stream) {
    (void)in_sizes; (void)n_in; (void)out_size; (void)d_ws; (void)ws_size;
    const float* x = (const float*)d_in[0];
    const float* y = (const float*)d_in[1];
    float* out = (float*)d_out;

    ssim_init_kernel<<<1, 1, 0, stream>>>(out);

    dim3 grid((IMG_W + TILE - 1) / TILE, (IMG_H + TILE - 1) / TILE, 3);  // 32x32x3
    size_t shmem = (size_t)(2 * P_ROWS * P_COLS + NQ * H_ROWS * H_STR) * sizeof(float);
    ssim_main_kernel<<<grid, 256, shmem, stream>>>(x, y, out);

    ssim_final_kernel<<<1, 1, 0, stream>>>(out);
}